// backWarp_MH_38104949850377
// MI455X (gfx1250) — compile-verified
//
#include <hip/hip_runtime.h>

// backWarp_MH: bilinear backward warp, B=4, C=3, H=W=256, NUM_OF=32 flows.
// Memory-bound gather op: ~160 MB streaming traffic -> ~7us floor at 23.3 TB/s.
// Strategy:
//   (1) repack img NCHW -> NHWC padded to 4ch (4 MB in d_ws): one b128 gather
//       per bilinear corner instead of 3 scattered b32 gathers.
//   (2) two x-pixels per thread: flow reads & output writes become b64 NT ops,
//       index decode amortized 2x.
//   (3) NT temporal hints on the 160 MB flow/out streams keep the 4 MB image
//       resident in WGP$/L2 (192 MB).

#define NOF 32
#define BB  4
#define CC  3
#define HH  256
#define WW  256
#define HWP (HH * WW)   // 65536

typedef float v2f __attribute__((ext_vector_type(2)));
typedef float v4f __attribute__((ext_vector_type(4)));

// ---- pass 1: img (B,C,H,W) -> imgT (B,H,W,4) with .w = 0 -------------------
__global__ __launch_bounds__(256) void repack_img_nhwc4(
    const float* __restrict__ img, v4f* __restrict__ imgT) {
  int idx = blockIdx.x * 256 + threadIdx.x;   // over B*H*W = 262144
  int p = idx & (HWP - 1);
  int b = idx >> 16;
  const float* s = img + (b * CC) * HWP + p;  // coalesced per-plane reads
  v4f t;
  t.x = s[0];
  t.y = s[HWP];
  t.z = s[2 * HWP];
  t.w = 0.0f;
  imgT[idx] = t;                              // coalesced b128 store, stays hot
}

// ---- pass 2 (fast path): warp using NHWC4 image, 2 pixels/thread -----------
__global__ __launch_bounds__(256) void warp_nhwc4_x2(
    const float* __restrict__ flow, const v4f* __restrict__ imgT,
    float* __restrict__ out) {
  int idx = blockIdx.x * 256 + threadIdx.x;   // over B*NOF*H*(W/2) = 4194304
  int x  = (idx & 127) << 1;                  // even x
  int y  = (idx >> 7) & 255;
  int of = (idx >> 15) & 31;
  int b  = idx >> 20;

  // flow[b, 2*of + {0,1}, y, x..x+1] : stream-once -> non-temporal b64 loads
  const v2f* fp = (const v2f*)(flow + ((b * 2 * NOF + 2 * of) * HWP + y * WW + x));
  v2f up = __builtin_nontemporal_load(fp);
  v2f vp = __builtin_nontemporal_load(fp + (HWP / 2));

  const v4f* base = imgT + b * HWP;
  float r[2][CC];

#pragma unroll
  for (int i = 0; i < 2; ++i) {
    float px = (float)(x + i) + up[i] - 0.5f;
    float py = (float)y + vp[i] - 0.5f;
    float x0f = floorf(px);
    float y0f = floorf(py);
    float wx = px - x0f;
    float wy = py - y0f;
    int x0 = (int)x0f, y0 = (int)y0f;
    int x1 = x0 + 1,   y1 = y0 + 1;

    int cx0 = min(max(x0, 0), WW - 1);
    int cx1 = min(max(x1, 0), WW - 1);
    int cy0 = min(max(y0, 0), HH - 1);
    int cy1 = min(max(y1, 0), HH - 1);

    bool vx0 = (unsigned)x0 < (unsigned)WW;
    bool vx1 = (unsigned)x1 < (unsigned)WW;
    bool vy0 = (unsigned)y0 < (unsigned)HH;
    bool vy1 = (unsigned)y1 < (unsigned)HH;

    // one b128 gather per bilinear corner (3 channels + pad)
    v4f v00 = base[cy0 * WW + cx0];
    v4f v01 = base[cy0 * WW + cx1];
    v4f v10 = base[cy1 * WW + cx0];
    v4f v11 = base[cy1 * WW + cx1];

    // zeros-padding folded into the weights (uniform control flow)
    float w00 = (vx0 && vy0) ? (1.0f - wx) * (1.0f - wy) : 0.0f;
    float w01 = (vx1 && vy0) ? wx * (1.0f - wy)          : 0.0f;
    float w10 = (vx0 && vy1) ? (1.0f - wx) * wy          : 0.0f;
    float w11 = (vx1 && vy1) ? wx * wy                   : 0.0f;

    r[i][0] = v00.x * w00 + v01.x * w01 + v10.x * w10 + v11.x * w11;
    r[i][1] = v00.y * w00 + v01.y * w01 + v10.y * w10 + v11.y * w11;
    r[i][2] = v00.z * w00 + v01.z * w01 + v10.z * w10 + v11.z * w11;
  }

  // out[b, of*3 + c, y, x..x+1] : write-once -> non-temporal b64 stores
  float* op = out + ((b * NOF + of) * CC * HWP + y * WW + x);
#pragma unroll
  for (int c = 0; c < CC; ++c) {
    v2f rr;
    rr.x = r[0][c];
    rr.y = r[1][c];
    __builtin_nontemporal_store(rr, (v2f*)(op + c * HWP));
  }
}

// ---- fallback: direct NCHW gathers (if d_ws too small) ---------------------
__global__ __launch_bounds__(256) void warp_nchw(
    const float* __restrict__ flow, const float* __restrict__ img,
    float* __restrict__ out) {
  int idx = blockIdx.x * 256 + threadIdx.x;   // over B*NOF*H*W = 8388608
  int x  = idx & (WW - 1);
  int y  = (idx >> 8) & (HH - 1);
  int of = (idx >> 16) & (NOF - 1);
  int b  = idx >> 21;

  const float* fp = flow + ((b * 2 * NOF + 2 * of) * HWP + y * WW + x);
  float u = __builtin_nontemporal_load(fp);
  float v = __builtin_nontemporal_load(fp + HWP);

  float px = (float)x + u - 0.5f;
  float py = (float)y + v - 0.5f;
  float x0f = floorf(px);
  float y0f = floorf(py);
  float wx = px - x0f;
  float wy = py - y0f;
  int x0 = (int)x0f, y0 = (int)y0f;
  int x1 = x0 + 1,   y1 = y0 + 1;

  int cx0 = min(max(x0, 0), WW - 1);
  int cx1 = min(max(x1, 0), WW - 1);
  int cy0 = min(max(y0, 0), HH - 1);
  int cy1 = min(max(y1, 0), HH - 1);

  bool vx0 = (unsigned)x0 < (unsigned)WW;
  bool vx1 = (unsigned)x1 < (unsigned)WW;
  bool vy0 = (unsigned)y0 < (unsigned)HH;
  bool vy1 = (unsigned)y1 < (unsigned)HH;

  float w00 = (vx0 && vy0) ? (1.0f - wx) * (1.0f - wy) : 0.0f;
  float w01 = (vx1 && vy0) ? wx * (1.0f - wy)          : 0.0f;
  float w10 = (vx0 && vy1) ? (1.0f - wx) * wy          : 0.0f;
  float w11 = (vx1 && vy1) ? wx * wy                   : 0.0f;

  int i00 = cy0 * WW + cx0;
  int i01 = cy0 * WW + cx1;
  int i10 = cy1 * WW + cx0;
  int i11 = cy1 * WW + cx1;

  const float* base = img + (b * CC) * HWP;
  float* op = out + ((b * NOF + of) * CC * HWP + y * WW + x);
#pragma unroll
  for (int c = 0; c < CC; ++c) {
    const float* pc = base + c * HWP;
    float r = pc[i00] * w00 + pc[i01] * w01 + pc[i10] * w10 + pc[i11] * w11;
    __builtin_nontemporal_store(r, op + c * HWP);
  }
}

extern "C" void kernel_launch(void* const* d_in, const int* in_sizes, int n_in,
                              void* d_out, int out_size, void* d_ws, size_t ws_size,
                              hipStream_t stream) {
  (void)in_sizes; (void)n_in; (void)out_size;
  const float* img  = (const float*)d_in[0];
  const float* flow = (const float*)d_in[1];
  float* out = (float*)d_out;

  const size_t need = (size_t)BB * HWP * sizeof(v4f);  // 4 MB

  if (ws_size >= need) {
    v4f* imgT = (v4f*)d_ws;
    repack_img_nhwc4<<<(BB * HWP) / 256, 256, 0, stream>>>(img, imgT);
    warp_nhwc4_x2<<<(BB * NOF * HWP / 2) / 256, 256, 0, stream>>>(flow, imgT, out);
  } else {
    warp_nchw<<<(BB * NOF * HWP) / 256, 256, 0, stream>>>(flow, img, out);
  }
}